// HungarianMatcher_42279658061976
// MI455X (gfx1250) — compile-verified
//
#include <hip/hip_runtime.h>
#include <hip/hip_bf16.h>
#include <math.h>

#define NB 50

typedef __attribute__((ext_vector_type(2))) float v2f;
typedef __attribute__((ext_vector_type(8))) float v8f;

// ---------------------------------------------------------------------------
// Rotated-box helpers (float32, mirrors reference semantics exactly)
// ---------------------------------------------------------------------------
__device__ __forceinline__ bool in_box(float ptx, float pty,
                                       float bx, float by, float bw, float bh,
                                       float ca, float sa) {
  float rx = ptx - bx, ry = pty - by;
  float lx = rx * ca + ry * sa;
  float ly = -rx * sa + ry * ca;
  return (fabsf(lx) <= 0.5f * bw + 1e-6f) && (fabsf(ly) <= 0.5f * bh + 1e-6f);
}

__device__ float giou3d_pair(const float* __restrict__ b1, const float* __restrict__ b2) {
  float x1 = b1[0], y1 = b1[1], w1 = b1[3], h1 = b1[4], a1 = b1[6];
  float x2 = b2[0], y2 = b2[1], w2 = b2[3], h2 = b2[4], a2 = b2[6];
  float ca1 = cosf(a1), sa1 = sinf(a1);
  float ca2 = cosf(a2), sa2 = sinf(a2);
  const float sxk[4] = {0.5f, -0.5f, -0.5f, 0.5f};
  const float syk[4] = {0.5f, 0.5f, -0.5f, -0.5f};
  float c1x[4], c1y[4], c2x[4], c2y[4];
  for (int k = 0; k < 4; ++k) {
    float xs = sxk[k] * w1, ys = syk[k] * h1;
    c1x[k] = xs * ca1 - ys * sa1 + x1;
    c1y[k] = xs * sa1 + ys * ca1 + y1;
    xs = sxk[k] * w2; ys = syk[k] * h2;
    c2x[k] = xs * ca2 - ys * sa2 + x2;
    c2y[k] = xs * sa2 + ys * ca2 + y2;
  }
  // 24 candidate points: 4 corners of each box + 16 edge-pair intersections
  float px[24], py[24]; int mk[24];
  for (int k = 0; k < 4; ++k) {
    px[k] = c1x[k]; py[k] = c1y[k];
    mk[k] = in_box(c1x[k], c1y[k], x2, y2, w2, h2, ca2, sa2) ? 1 : 0;
    px[4 + k] = c2x[k]; py[4 + k] = c2y[k];
    mk[4 + k] = in_box(c2x[k], c2y[k], x1, y1, w1, h1, ca1, sa1) ? 1 : 0;
  }
  int idx = 8;
  for (int e1 = 0; e1 < 4; ++e1) {
    float p1x = c1x[e1], p1y = c1y[e1];
    float d1x = c1x[(e1 + 1) & 3] - p1x, d1y = c1y[(e1 + 1) & 3] - p1y;
    for (int e2 = 0; e2 < 4; ++e2) {
      float q1x = c2x[e2], q1y = c2y[e2];
      float d2x = c2x[(e2 + 1) & 3] - q1x, d2y = c2y[(e2 + 1) & 3] - q1y;
      float dqx = q1x - p1x, dqy = q1y - p1y;
      float den = d1x * d2y - d1y * d2x;
      float dens = (fabsf(den) < 1e-9f) ? 1.0f : den;
      float t = (dqx * d2y - dqy * d2x) / dens;
      float uu = (dqx * d1y - dqy * d1x) / dens;
      mk[idx] = ((fabsf(den) > 1e-9f) && t >= 0.0f && t <= 1.0f &&
                 uu >= 0.0f && uu <= 1.0f) ? 1 : 0;
      px[idx] = p1x + t * d1x;
      py[idx] = p1y + t * d1y;
      ++idx;
    }
  }
  float kf = 0.0f, sx = 0.0f, sy = 0.0f;
  for (int k = 0; k < 24; ++k) {
    float m = (float)mk[k];
    kf += m; sx += px[k] * m; sy += py[k] * m;
  }
  float cx = sx / fmaxf(kf, 1.0f);
  float cy = sy / fmaxf(kf, 1.0f);
  float ang[24]; int ord[24];
  for (int k = 0; k < 24; ++k) {
    ang[k] = mk[k] ? atan2f(py[k] - cy, px[k] - cx) : 1.0e9f;
    ord[k] = k;
  }
  // stable insertion sort (matches jnp stable argsort)
  for (int i = 1; i < 24; ++i) {
    int oi = ord[i]; float ai = ang[oi];
    int j = i - 1;
    while (j >= 0 && ang[ord[j]] > ai) { ord[j + 1] = ord[j]; --j; }
    ord[j + 1] = oi;
  }
  int kc = (int)kf;
  float vx[24], vy[24];
  for (int k = 0; k < 24; ++k) {
    int o = ord[k];
    float m = (float)mk[o];
    vx[k] = (px[o] - cx) * m;
    vy[k] = (py[o] - cy) * m;
  }
  float ssum = 0.0f;
  for (int k = 0; k < kc; ++k) {
    int nx = (k + 1 < kc) ? k + 1 : 0;
    ssum += vx[k] * vy[nx] - vy[k] * vx[nx];
  }
  float inter2d = 0.5f * fabsf(ssum);

  float zmax1 = b1[2] + 0.5f * b1[5], zmin1 = b1[2] - 0.5f * b1[5];
  float zmax2 = b2[2] + 0.5f * b2[5], zmin2 = b2[2] - 0.5f * b2[5];
  float z_ov  = fmaxf(0.0f, fminf(zmax1, zmax2) - fmaxf(zmin1, zmin2));
  float z_rng = fmaxf(0.0f, fmaxf(zmax1, zmax2) - fminf(zmin1, zmin2));
  float inter3d = inter2d * z_ov;
  float vol1 = b1[3] * b1[4] * b1[5];
  float vol2 = b2[3] * b2[4] * b2[5];
  float u3d = vol1 + vol2 - inter3d;
  float xmn = c1x[0], xmx = c1x[0], ymn = c1y[0], ymx = c1y[0];
  for (int k = 1; k < 4; ++k) {
    xmn = fminf(xmn, c1x[k]); xmx = fmaxf(xmx, c1x[k]);
    ymn = fminf(ymn, c1y[k]); ymx = fmaxf(ymx, c1y[k]);
  }
  for (int k = 0; k < 4; ++k) {
    xmn = fminf(xmn, c2x[k]); xmx = fmaxf(xmx, c2x[k]);
    ymn = fminf(ymn, c2y[k]); ymx = fmaxf(ymx, c2y[k]);
  }
  float v_c = z_rng * (xmx - xmn) * (ymx - ymn);
  float iou3d = inter3d / u3d;
  return 1.0f - iou3d + (v_c - u3d) / v_c;
}

// ---------------------------------------------------------------------------
// Kernel A: one thread per (pred, target) pair -> box GIoU cost
// ---------------------------------------------------------------------------
__global__ __launch_bounds__(128) void cost_box_kernel(const float* __restrict__ predBox,
                                                       const float* __restrict__ tgtBox,
                                                       float* __restrict__ boxC) {
  int idx = blockIdx.x * blockDim.x + threadIdx.x;
  if (idx >= NB * NB) return;
  int i = idx / NB, j = idx % NB;
  __builtin_prefetch(predBox + i * 7, 0, 0);   // gfx1250 global_prefetch_b8
  __builtin_prefetch(tgtBox + j * 7, 0, 0);
  boxC[idx] = giou3d_pair(predBox + i * 7, tgtBox + j * 7);
}

// ---------------------------------------------------------------------------
// Kernel B: single wave32. Score-cost via V_WMMA_F32_16X16X4_F32 outer
// products, cost assembly in LDS, JV Hungarian with wave-parallel delta
// scan (2 columns/lane), final matched sums.
// ---------------------------------------------------------------------------
__global__ __launch_bounds__(32) void match_kernel(const float* __restrict__ boxC,
                                                   const float* __restrict__ pscore,
                                                   const float* __restrict__ tscore,
                                                   float* __restrict__ out) {
  __shared__ float sCost[NB * NB];
  __shared__ float sScore[NB * NB];
  __shared__ float sBox[NB * NB];
  __shared__ double uRow[NB + 1];
  __shared__ int pCol[64];
  __shared__ int wayCol[64];
  __shared__ float sPS[64], sTS[64];

  const int lane = threadIdx.x;

  for (int k = lane; k < 64; k += 32) {
    sPS[k] = (k < NB) ? pscore[k] : 0.0f;
    sTS[k] = (k < NB) ? tscore[k] : 0.0f;
    pCol[k] = 0; wayCol[k] = 0;
  }
  for (int k = lane; k <= NB; k += 32) uRow[k] = 0.0;
  __threadfence_block();

  // score_cost[i][j] = relu(t[j] - p[i]) = relu( [1, -p] x [t; 1] )
  // A (16x4 f32): A[:,0]=1, A[:,1]=-p  -> lanes 0-15 hold K=0,1; lanes 16-31 (K=2,3)=0
  // B (4x16 f32): B[0,:]=t, B[1,:]=1   -> same half-wave striping
  const int mrow = lane & 15;
  const bool lo = lane < 16;
  for (int ti = 0; ti < 4; ++ti) {
    float pv = sPS[ti * 16 + mrow];
    v2f a;
    a.x = lo ? 1.0f : 0.0f;
    a.y = lo ? -pv : 0.0f;
    for (int tj = 0; tj < 4; ++tj) {
      float tv = sTS[tj * 16 + mrow];
      v2f b;
      b.x = lo ? tv : 0.0f;
      b.y = lo ? 1.0f : 0.0f;
      v8f c = {};
      v8f d = __builtin_amdgcn_wmma_f32_16x16x4_f32(
          false, a, false, b, (short)0, c, false, false);
      int nCol = tj * 16 + (lane & 15);
      int mBase = ti * 16 + ((lane >= 16) ? 8 : 0);
      for (int r = 0; r < 8; ++r) {
        int gi = mBase + r;
        if (gi < NB && nCol < NB) {
          float sc = fmaxf(0.0f, d[r]);
          float bc = boxC[gi * NB + nCol];
          sScore[gi * NB + nCol] = sc;
          sBox[gi * NB + nCol]   = bc;
          sCost[gi * NB + nCol]  = sc + bc;
        }
      }
    }
  }
  __threadfence_block();

  // ---- JV Hungarian, float64 potentials (matches reference) ----
  volatile double* vu = uRow;
  volatile int* vp = pCol;
  volatile int* vw = wayCol;
  const double INF = 1.0e18;
  const int jA = lane;        // column slot 0 (j==0 is the dummy column)
  const int jB = lane + 32;   // column slot 1
  const bool vaA = (jA >= 1) && (jA <= NB);
  const bool vaB = (jB <= NB);
  double vA = 0.0, vB = 0.0;  // column potentials v[jA], v[jB]

  for (int i = 1; i <= NB; ++i) {
    if (lane == 0) vp[0] = i;
    __threadfence_block();
    double mA = INF, mB = INF;
    bool uA = false, uB = false;
    int j0 = 0;
    while (true) {
      if (jA == j0) uA = true;
      if (jB == j0) uB = true;
      int i0 = vp[j0];
      double u0 = vu[i0];
      if (vaA && !uA) {
        double cur = (double)sCost[(i0 - 1) * NB + (jA - 1)] - u0 - vA;
        if (cur < mA) { mA = cur; vw[jA] = j0; }
      }
      if (vaB && !uB) {
        double cur = (double)sCost[(i0 - 1) * NB + (jB - 1)] - u0 - vB;
        if (cur < mB) { mB = cur; vw[jB] = j0; }
      }
      // wave-wide argmin over unused columns, tie-break to smallest j
      double bv = INF; int bj = 0x7FFFFFFF;
      if (vaA && !uA) { bv = mA; bj = jA; }
      if (vaB && !uB) {
        if (mB < bv || (mB == bv && jB < bj)) { bv = mB; bj = jB; }
      }
      for (int off = 16; off > 0; off >>= 1) {
        double ov = __shfl_xor(bv, off, 32);
        int oj = __shfl_xor(bj, off, 32);
        if (ov < bv || (ov == bv && oj < bj)) { bv = ov; bj = oj; }
      }
      double delta = bv;
      int j1 = bj;
      if (jA <= NB) {
        if (uA) { int r = vp[jA]; vu[r] = vu[r] + delta; vA -= delta; }
        else if (vaA) mA -= delta;
      }
      if (jB <= NB) {
        if (uB) { int r = vp[jB]; vu[r] = vu[r] + delta; vB -= delta; }
        else mB -= delta;
      }
      __threadfence_block();
      j0 = j1;
      if (vp[j0] == 0) break;
    }
    // augmenting path (cheap, sequential on lane 0)
    if (lane == 0) {
      int jj = j0;
      while (true) {
        int jpp = vw[jj];
        vp[jj] = vp[jpp];
        jj = jpp;
        if (jj == 0) break;
      }
    }
    __threadfence_block();
  }

  // ---- matched sums ----
  double accC = 0.0, accS = 0.0, accB = 0.0;
  for (int s = 0; s < 2; ++s) {
    int j = lane + 32 * s;
    if (j >= 1 && j <= NB) {
      int pr = vp[j];
      if (pr > 0) {
        int off = (pr - 1) * NB + (j - 1);
        accC += (double)sCost[off];
        accS += (double)sScore[off];
        accB += (double)sBox[off];
      }
    }
  }
  for (int off = 16; off > 0; off >>= 1) {
    accC += __shfl_xor(accC, off, 32);
    accS += __shfl_xor(accS, off, 32);
    accB += __shfl_xor(accB, off, 32);
  }
  if (lane == 0) {
    out[0] = (float)(accC / (double)NB);
    out[1] = (float)(accS / (double)NB);
    out[2] = (float)(accB / (double)NB);
  }
}

// ---------------------------------------------------------------------------
extern "C" void kernel_launch(void* const* d_in, const int* in_sizes, int n_in,
                              void* d_out, int out_size, void* d_ws, size_t ws_size,
                              hipStream_t stream) {
  (void)in_sizes; (void)n_in; (void)out_size; (void)ws_size;
  const float* predBox = (const float*)d_in[0];
  const float* pscore  = (const float*)d_in[1];
  const float* tgtBox  = (const float*)d_in[2];
  const float* tscore  = (const float*)d_in[3];
  float* out  = (float*)d_out;
  float* boxC = (float*)d_ws;   // NB*NB floats of scratch

  cost_box_kernel<<<(NB * NB + 127) / 128, 128, 0, stream>>>(predBox, tgtBox, boxC);
  match_kernel<<<1, 32, 0, stream>>>(boxC, pscore, tscore, out);
}